// BackpropWiSARD_22952305230076
// MI455X (gfx1250) — compile-verified
//
#include <hip/hip_runtime.h>
#include <hip/hip_bf16.h>
#include <stdint.h>

#define B_   256
#define NI_  1024
#define C_   100
#define U_   16
#define E_   2048
#define H_   4
#define BPI_ 8
#define IB_  (NI_*BPI_)   // 8192
#define F_   (IB_/U_)     // 512

typedef __attribute__((ext_vector_type(2))) float v2f;
typedef __attribute__((ext_vector_type(4))) float v4f;
typedef __attribute__((ext_vector_type(8))) float v8f;

// ---------------------------------------------------------------------------
// Kernel A: pack sign bits of data (C,F,E) f32 -> packed[f][e][4 x u32],
// bit c of word (c>>5) = (data[c,f,e] >= 0).  Reads all 419 MB once,
// coalesced (b128 per lane).  Loads are NON-TEMPORAL: the 419 MB stream is
// single-use and must not evict the 16.8 MB packed table (+ maskT) that the
// gather pass needs resident in the 192 MB L2.  ~18 us @ 23.3 TB/s HBM.
// Fully unrolled over c so the word index (c>>5) is compile-time -> registers.
// ---------------------------------------------------------------------------
__global__ __launch_bounds__(256) void pack_signs(const float* __restrict__ data,
                                                  uint32_t* __restrict__ packed) {
  int tid = blockIdx.x * blockDim.x + threadIdx.x;   // [0, F*E/4)
  int f  = tid / (E_ / 4);
  int e4 = tid % (E_ / 4);
  int e  = e4 * 4;

  uint32_t w0[4] = {0,0,0,0};   // words for e+0
  uint32_t w1[4] = {0,0,0,0};   // e+1
  uint32_t w2[4] = {0,0,0,0};   // e+2
  uint32_t w3[4] = {0,0,0,0};   // e+3

  const v4f* dp = (const v4f*)data;
  #pragma unroll
  for (int c = 0; c < C_; ++c) {
    v4f v = __builtin_nontemporal_load(&dp[(size_t)(c * F_ + f) * (E_ / 4) + e4]);
    const int      ws  = c >> 5;          // compile-time after unroll
    const uint32_t bit = 1u << (c & 31);
    if (v[0] >= 0.0f) w0[ws] |= bit;
    if (v[1] >= 0.0f) w1[ws] |= bit;
    if (v[2] >= 0.0f) w2[ws] |= bit;
    if (v[3] >= 0.0f) w3[ws] |= bit;
  }

  uint4* pp = (uint4*)packed;             // regular (RT) stores: keep in L2
  size_t base = (size_t)f * E_ + e;
  pp[base + 0] = make_uint4(w0[0], w0[1], w0[2], w0[3]);
  pp[base + 1] = make_uint4(w1[0], w1[1], w1[2], w1[3]);
  pp[base + 2] = make_uint4(w2[0], w2[1], w2[2], w2[3]);
  pp[base + 3] = make_uint4(w3[0], w3[1], w3[2], w3[3]);
}

// ---------------------------------------------------------------------------
// Kernel A2: maskT[f][c] = mask[c][f] so the per-class reduction loop reads
// contiguous (coalesced, L2-resident 200 KB) rows.  Source reads are
// single-use -> non-temporal.
// ---------------------------------------------------------------------------
__global__ __launch_bounds__(256) void transpose_mask(const float* __restrict__ mask,
                                                      float* __restrict__ maskT) {
  int t = blockIdx.x * blockDim.x + threadIdx.x;
  if (t < F_ * C_) {
    int f = t / C_, c = t % C_;
    maskT[t] = __builtin_nontemporal_load(&mask[c * F_ + f]);
  }
}

// ---------------------------------------------------------------------------
// Kernel B: one block per batch sample.
//  Phase 1: per filter f, compute the 4 XOR-hash indices from LDS-staged
//           x-row / thresholds / permutation, then AND the four 128-bit
//           class masks gathered (16 B each) from the L2-resident packed
//           table -> cmask_lds[f].   (min over h of 0/1 == bitwise AND)
//  Phase 2: thread c reduces over f: acc += bit(cmask[f],c) ? maskT[f][c].
//           cmask word is an LDS broadcast (all lanes of a 32-group share it).
// Writes partial (un-biased) output; bias added by the WMMA pass.
// ---------------------------------------------------------------------------
__global__ __launch_bounds__(128) void wisard_main(const float* __restrict__ x,
                                                   const float* __restrict__ thr,
                                                   const int*   __restrict__ hashv,
                                                   const int*   __restrict__ order,
                                                   const uint32_t* __restrict__ packed,
                                                   const float* __restrict__ maskT,
                                                   float* __restrict__ out) {
  __shared__ float    x_lds[NI_];          //  4 KB
  __shared__ float    thr_lds[NI_ * BPI_]; // 32 KB
  __shared__ uint16_t ord_lds[IB_];        // 16 KB (values < 8192)
  __shared__ int      hv_lds[H_ * U_];     // 256 B
  __shared__ uint4    cmask_lds[F_];       //  8 KB

  const int b   = blockIdx.x;
  const int tid = threadIdx.x;             // blockDim = 128 (4 waves)

  for (int i = tid; i < NI_; i += 128)        x_lds[i]  = x[b * NI_ + i];
  for (int i = tid; i < NI_ * BPI_; i += 128) thr_lds[i] = thr[i];
  for (int i = tid; i < IB_; i += 128)        ord_lds[i] = (uint16_t)order[i];
  if (tid < H_ * U_)                          hv_lds[tid] = hashv[tid];
  __syncthreads();

  // ---- Phase 1: hashes + AND-gather of class masks ----
  for (int f = tid; f < F_; f += 128) {
    int i0 = 0, i1 = 0, i2 = 0, i3 = 0;
    #pragma unroll
    for (int u = 0; u < U_; ++u) {
      int k = (int)ord_lds[f * U_ + u];
      // bits[b, k] = x[b, k>>3] >= thresholds[k>>3][k&7]  (linear index k)
      int m = (x_lds[k >> 3] >= thr_lds[k]) ? -1 : 0;
      i0 ^= hv_lds[0 * U_ + u] & m;
      i1 ^= hv_lds[1 * U_ + u] & m;
      i2 ^= hv_lds[2 * U_ + u] & m;
      i3 ^= hv_lds[3 * U_ + u] & m;
    }
    i0 &= (E_ - 1); i1 &= (E_ - 1); i2 &= (E_ - 1); i3 &= (E_ - 1);

    const uint4* pp = (const uint4*)packed;
    size_t fb = (size_t)f * E_;
    uint4 a  = pp[fb + i0];
    uint4 bb = pp[fb + i1];
    uint4 c  = pp[fb + i2];
    uint4 d  = pp[fb + i3];
    uint4 r;
    r.x = a.x & bb.x & c.x & d.x;
    r.y = a.y & bb.y & c.y & d.y;
    r.z = a.z & bb.z & c.z & d.z;
    r.w = a.w & bb.w & c.w & d.w;
    cmask_lds[f] = r;
  }
  __syncthreads();

  // ---- Phase 2: per-class reduction over filters ----
  if (tid < C_) {
    const int c = tid;
    const int ws = c >> 5;
    const uint32_t bit = 1u << (c & 31);
    float acc = 0.0f;
    #pragma unroll 4
    for (int f = 0; f < F_; ++f) {
      const uint32_t* cw = (const uint32_t*)&cmask_lds[f];
      float mv = maskT[f * C_ + c];                  // coalesced, L2-hot
      acc += (cw[ws] & bit) ? mv : 0.0f;             // v_cndmask + v_add
    }
    out[b * C_ + c] = acc;                           // bias added by WMMA pass
  }
}

// ---------------------------------------------------------------------------
// Kernel C: bias add performed on the matrix pipe, exactly:
//   D = A(16x4) x B(4x16) + C,  A row = (1,0,0,0), B row0 = bias  =>  D = C + bias.
// f32 WMMA, so the math is bit-exact. One wave per 16x16 output tile.
// A layout (ISA 7.12.2): lanes 0-15 VGPR0 hold K=0 -> set 1.0 there, 0 elsewhere.
// C/D layout: VGPR r, lane l -> M = r + 8*(l>=16), N = l%16.
// ---------------------------------------------------------------------------
__global__ __launch_bounds__(32) void bias_wmma(float* __restrict__ out,
                                                const float* __restrict__ bias) {
  const int tile = blockIdx.x;        // 16 b-tiles * 7 c-tiles
  const int tb   = tile / 7;
  const int tc   = tile % 7;
  const int lane = threadIdx.x;       // 0..31, wave32
  const int n    = lane & 15;
  const int half = lane >> 4;

  const int col  = tc * 16 + n;
  const int colc = (col < C_) ? col : (C_ - 1);     // clamped, no EXEC change
  const float bval = (col < C_) ? bias[colc] : 0.0f;

  v8f cacc;
  #pragma unroll
  for (int r = 0; r < 8; ++r) {
    int row = tb * 16 + r + 8 * half;               // always < B_
    float v = out[row * C_ + colc];
    cacc[r] = (col < C_) ? v : 0.0f;
  }

  v2f aM; aM[0] = (half == 0) ? 1.0f : 0.0f; aM[1] = 0.0f;   // A[m,k] = (k==0)
  v2f bM; bM[0] = bval; bM[1] = bval;                         // B[k,n] = bias[n]

  // 8 args: (neg_a, A, neg_b, B, c_mod, C, reuse_a, reuse_b)
  v8f d = __builtin_amdgcn_wmma_f32_16x16x4_f32(false, aM, false, bM,
                                                (short)0, cacc, false, false);

  #pragma unroll
  for (int r = 0; r < 8; ++r) {
    int row = tb * 16 + r + 8 * half;
    if (col < C_) out[row * C_ + col] = d[r];
  }
}

// ---------------------------------------------------------------------------
extern "C" void kernel_launch(void* const* d_in, const int* in_sizes, int n_in,
                              void* d_out, int out_size, void* d_ws, size_t ws_size,
                              hipStream_t stream) {
  (void)in_sizes; (void)n_in; (void)out_size; (void)ws_size;
  const float* x     = (const float*)d_in[0];
  const float* thr   = (const float*)d_in[1];
  const float* data  = (const float*)d_in[2];
  const int*   hashv = (const int*)d_in[3];
  const int*   order = (const int*)d_in[4];
  const float* mask  = (const float*)d_in[5];
  const float* bias  = (const float*)d_in[6];
  float* out = (float*)d_out;

  // Workspace layout: 16 MB packed sign table + 200 KB transposed mask.
  uint32_t* packed = (uint32_t*)d_ws;
  float*    maskT  = (float*)((char*)d_ws + (size_t)F_ * E_ * 4 * sizeof(uint32_t));

  pack_signs<<<(F_ * E_ / 4) / 256, 256, 0, stream>>>(data, packed);
  transpose_mask<<<(F_ * C_ + 255) / 256, 256, 0, stream>>>(mask, maskT);
  wisard_main<<<B_, 128, 0, stream>>>(x, thr, hashv, order, packed, maskT, out);
  bias_wmma<<<(B_ / 16) * ((C_ + 15) / 16), 32, 0, stream>>>(out, bias);
}